// PointPillarsScatter_66889820668082
// MI455X (gfx1250) — compile-verified
//
#include <hip/hip_runtime.h>

// PointPillarsScatter for MI455X (gfx1250, wave32).
// Bandwidth-bound scatter -> reformulated as a single coalesced gather pass:
//   pass 1: map[b*ny*nx + y*nx + x] = point_id   (4 MB scratch, -1 = empty)
//   pass 2: out[b,c,y,x] = map>=0 ? feat[pid,c] : 0, written as b128 NT stores
// Traffic ~290 MB vs ~700+ MB for zero-fill+scatter; @23.3 TB/s => ~12.5 us.

typedef __attribute__((ext_vector_type(4))) float floatx4;
typedef __attribute__((ext_vector_type(4))) int   intx4;

// ---------------------------------------------------------------- pass 1a: map = -1
__global__ void pps_init_map(int* __restrict__ map, int n4) {
    const int stride = gridDim.x * blockDim.x;
    const intx4 m1 = {-1, -1, -1, -1};
    for (int i = blockIdx.x * blockDim.x + threadIdx.x; i < n4; i += stride) {
        ((intx4*)map)[i] = m1;            // regular stores: map is re-read from L2 next kernel
    }
}

// ---------------------------------------------------------------- pass 1b: scatter ids
__global__ void pps_scatter_ids(const int* __restrict__ coords, int* __restrict__ map,
                                int P, int plane, int nx) {
    const int p = blockIdx.x * blockDim.x + threadIdx.x;
    if (p < P) {
        const int b = coords[p * 4 + 0];
        const int y = coords[p * 4 + 2];
        const int x = coords[p * 4 + 3];
        map[b * plane + y * nx + x] = p;  // cells unique per batch -> race-free
    }
}

// ---------------------------------------------------------------- pass 2: gather + NT store
// One thread per 4-wide x-quad; loops over all C channels so the 4 map entries
// are loaded once and the feature rows stream as 16B vector loads from L2.
__global__ void pps_gather(const float* __restrict__ feat, const int* __restrict__ map,
                           float* __restrict__ out, int C, int plane, int quadsTotal) {
    const int t = blockIdx.x * blockDim.x + threadIdx.x;
    if (t >= quadsTotal) return;

    const int quadsPerImg = plane >> 2;
    const int b = t / quadsPerImg;
    const int q = t - b * quadsPerImg;

    const intx4 pid = ((const intx4*)map)[t];   // map is [B*plane] so quad index == t

    float* outBase = out + (size_t)b * (size_t)C * (size_t)plane + (size_t)(q << 2);
    const floatx4 zero = {0.f, 0.f, 0.f, 0.f};

    const bool v0 = pid.x >= 0, v1 = pid.y >= 0, v2 = pid.z >= 0, v3 = pid.w >= 0;

    if (!(v0 | v1 | v2 | v3)) {
        // ~67% of quads: pure zero stream, no gathers at all.
        #pragma unroll 8
        for (int c = 0; c < C; ++c) {
            __builtin_nontemporal_store(zero, (floatx4*)(outBase + (size_t)c * plane));
        }
        return;
    }

    const float* f0 = feat + (long)pid.x * C;
    const float* f1 = feat + (long)pid.y * C;
    const float* f2 = feat + (long)pid.z * C;
    const float* f3 = feat + (long)pid.w * C;

    if ((C & 3) == 0) {
        // Vectorized: 4 channels per step -> 4x b128 gathers + 4x b128 NT stores.
        // feat rows are 16B-aligned when C%4==0 (C=64 here -> 256B aligned).
        for (int cg = 0; cg < C; cg += 4) {
            floatx4 a = v0 ? *(const floatx4*)(f0 + cg) : zero;
            floatx4 bq = v1 ? *(const floatx4*)(f1 + cg) : zero;
            floatx4 cq = v2 ? *(const floatx4*)(f2 + cg) : zero;
            floatx4 dq = v3 ? *(const floatx4*)(f3 + cg) : zero;
            // 4x4 transpose: output channel k gets lane-k of each row quad
            floatx4 o0 = {a.x, bq.x, cq.x, dq.x};
            floatx4 o1 = {a.y, bq.y, cq.y, dq.y};
            floatx4 o2 = {a.z, bq.z, cq.z, dq.z};
            floatx4 o3 = {a.w, bq.w, cq.w, dq.w};
            __builtin_nontemporal_store(o0, (floatx4*)(outBase + (size_t)(cg + 0) * plane));
            __builtin_nontemporal_store(o1, (floatx4*)(outBase + (size_t)(cg + 1) * plane));
            __builtin_nontemporal_store(o2, (floatx4*)(outBase + (size_t)(cg + 2) * plane));
            __builtin_nontemporal_store(o3, (floatx4*)(outBase + (size_t)(cg + 3) * plane));
        }
    } else {
        for (int c = 0; c < C; ++c) {
            floatx4 v;
            v.x = v0 ? f0[c] : 0.f;
            v.y = v1 ? f1[c] : 0.f;
            v.z = v2 ? f2[c] : 0.f;
            v.w = v3 ? f3[c] : 0.f;
            __builtin_nontemporal_store(v, (floatx4*)(outBase + (size_t)c * plane));
        }
    }
}

// ---------------------------------------------------------------- fallback (ws too small)
__global__ void pps_zero_canvas(float* __restrict__ out, int n4) {
    const int stride = gridDim.x * blockDim.x;
    const floatx4 z = {0.f, 0.f, 0.f, 0.f};
    for (int i = blockIdx.x * blockDim.x + threadIdx.x; i < n4; i += stride) {
        __builtin_nontemporal_store(z, ((floatx4*)out) + i);
    }
}

__global__ void pps_scatter_direct(const float* __restrict__ feat, const int* __restrict__ coords,
                                   float* __restrict__ out, int P, int C, int plane, int nx) {
    const int t = blockIdx.x * blockDim.x + threadIdx.x;
    const int p = t / C;
    const int c = t - p * C;
    if (p < P) {
        const int b = coords[p * 4 + 0];
        const int y = coords[p * 4 + 2];
        const int x = coords[p * 4 + 3];
        out[((size_t)(b * C + c)) * plane + (size_t)(y * nx + x)] = feat[(size_t)p * C + c];
    }
}

// ---------------------------------------------------------------- launch
extern "C" void kernel_launch(void* const* d_in, const int* in_sizes, int n_in,
                              void* d_out, int out_size, void* d_ws, size_t ws_size,
                              hipStream_t stream) {
    const float* feat   = (const float*)d_in[0];   // [P, C] fp32
    const int*   coords = (const int*)d_in[1];     // [P, 4] int32 (b, z, y, x)
    float*       out    = (float*)d_out;           // [B, C, ny, nx] fp32

    // Derive shapes from sizes; spatial dims are fixed by the reference.
    const int P  = in_sizes[1] / 4;                // 100000
    const int C  = (P > 0) ? in_sizes[0] / P : 64; // 64
    const int NX = 512, NY = 512;
    const int plane = NX * NY;                     // 262144
    const int B  = (C > 0) ? out_size / (C * plane) : 4;  // 4

    const int threads = 256;                       // 8 wave32 waves per block
    const size_t mapBytes = (size_t)B * (size_t)plane * sizeof(int);   // 4 MB

    if (ws_size >= mapBytes) {
        int* map = (int*)d_ws;
        const int mapQuads = (B * plane) >> 2;     // 262144
        pps_init_map<<<(mapQuads + threads - 1) / threads, threads, 0, stream>>>(map, mapQuads);
        pps_scatter_ids<<<(P + threads - 1) / threads, threads, 0, stream>>>(coords, map, P, plane, NX);
        const int quads = (B * plane) >> 2;        // 262144 threads, each writes C channels
        pps_gather<<<(quads + threads - 1) / threads, threads, 0, stream>>>(feat, map, out, C, plane, quads);
    } else {
        // Deterministic fallback: zero-fill then direct scatter.
        const int n4 = out_size >> 2;              // 16.7M float4 quads
        pps_zero_canvas<<<2048, threads, 0, stream>>>(out, n4);
        const int total = P * C;
        pps_scatter_direct<<<(total + threads - 1) / threads, threads, 0, stream>>>(
            feat, coords, out, P, C, plane, NX);
    }
}